// GAT_55602646614063
// MI455X (gfx1250) — compile-verified
//
#include <hip/hip_runtime.h>
#include <hip/hip_bf16.h>

// GAT on MI455X (gfx1250), compile-only tuning.
//
// Exact algebraic restructuring (no approximation):
//   out[n] = sum_h  num[n,h] / (den[n,h] + 1e-16)  +  (fc_w . bias + fc_b)
// where per-node scalars a_src/a_dst/p come from x @ W2, W2 = [F_IN x 12]
// pre-folded from lin_w/att_src/att_dst/fc_w. Edge phase touches only 4
// scalars per edge instead of 256 floats.
//
// ws requirement: ~9.7 MB (asserted-by-construction below).

#define N_NODES 100000
#define N_EDGES 1600000
#define F_IN    256
#define HEADS   4
#define HIDDEN  64
#define NEG_SLOPE 0.2f

typedef __attribute__((ext_vector_type(2))) float v2f;
typedef __attribute__((ext_vector_type(4))) float v4f;
typedef __attribute__((ext_vector_type(8))) float v8f;

// ---- workspace layout (units of 4 bytes) --------------------------------
#define OFF_W2   0                         // 256*16 folded weights (cols 12..15 zero)
#define OFF_CC   (OFF_W2 + F_IN * 16)      // 1 scalar: fc_w.bias + fc_b
#define OFF_ASRC (OFF_CC + 16)             // N*4
#define OFF_ADST (OFF_ASRC + N_NODES * 4)  // N*4
#define OFF_P    (OFF_ADST + N_NODES * 4)  // N*4
#define OFF_MU   (OFF_P    + N_NODES * 4)  // N*4 (uint, orderable-encoded max)
#define OFF_DEN  (OFF_MU   + N_NODES * 4)  // N*4
#define OFF_NUM  (OFF_DEN  + N_NODES * 4)  // N*4

__device__ __forceinline__ unsigned fp32_orderable(float f) {
    unsigned u = __float_as_uint(f);
    return (u & 0x80000000u) ? ~u : (u | 0x80000000u);
}
__device__ __forceinline__ float fp32_unorderable(unsigned u) {
    return __uint_as_float((u & 0x80000000u) ? (u & 0x7fffffffu) : ~u);
}
__device__ __forceinline__ float leaky(float e) {
    return fmaxf(e, NEG_SLOPE * e);
}

// ---- 0) fold weights: W2[c][j], j = h*3+t, t in {att_src, att_dst, fc} --
__global__ void gat_fold_w(const float* __restrict__ lin_w,
                           const float* __restrict__ att_src,
                           const float* __restrict__ att_dst,
                           const float* __restrict__ bias,
                           const float* __restrict__ fc_w,
                           const float* __restrict__ fc_b,
                           float* __restrict__ ws) {
    int c = threadIdx.x;  // 0..255, input-feature index
    float* W2 = ws + OFF_W2;
    for (int j = 0; j < 16; ++j) {
        float s = 0.f;
        if (j < 12) {
            int h = j / 3, t = j - 3 * h;
            const float* coef = (t == 0) ? att_src : (t == 1) ? att_dst : fc_w;
            const float* lw = lin_w + (h * HIDDEN) * F_IN + c;  // lin_w[h*64+c2][c]
            const float* cf = coef + h * HIDDEN;
#pragma unroll 8
            for (int c2 = 0; c2 < HIDDEN; ++c2)
                s = fmaf(cf[c2], lw[c2 * F_IN], s);
        }
        W2[c * 16 + j] = s;
    }
    if (c == 0) {
        float cc = fc_b[0];
        for (int j = 0; j < HEADS * HIDDEN; ++j) cc = fmaf(fc_w[j], bias[j], cc);
        ws[OFF_CC] = cc;
    }
}

// ---- 1) init per-node accumulators --------------------------------------
__global__ void gat_init(float* __restrict__ ws) {
    int i = blockIdx.x * blockDim.x + threadIdx.x;
    if (i < N_NODES * 4) {
        ((unsigned*)(ws + OFF_MU))[i] = fp32_orderable(-__builtin_huge_valf());
        (ws + OFF_DEN)[i] = 0.f;
        (ws + OFF_NUM)[i] = 0.f;
    }
}

// ---- 2) node pass: [a_src a_dst p] = x @ W2 via V_WMMA_F32_16X16X4_F32 --
// One wave handles a 16-node tile: M=16 nodes, N=16 cols (12 used), K=256.
__global__ void __launch_bounds__(128)
gat_proj_wmma(const float* __restrict__ x, float* __restrict__ ws) {
    const int NT = N_NODES / 16;  // 6250 tiles exactly
    int tile = blockIdx.x * 4 + (threadIdx.x >> 5);
    if (tile >= NT) return;       // wave-uniform; EXEC stays full for WMMA

    int lane = threadIdx.x & 31;
    int half = lane >> 4;         // 0: K pair {0,1}; 1: K pair {2,3}
    int lr   = lane & 15;

    const float* xrow = x + (tile * 16 + lr) * F_IN;  // A row for this lane
    const float* W2   = ws + OFF_W2;

    v8f acc = {};
#pragma unroll 4
    for (int k = 0; k < F_IN; k += 4) {
        // A 16x4 f32 (ISA 7.12.2): lanes 0-15 rows M=0..15, V0=K0 V1=K1;
        // lanes 16-31 same rows, V0=K2 V1=K3.
        v2f a;
        a[0] = xrow[k + 2 * half + 0];
        a[1] = xrow[k + 2 * half + 1];
        // B 4x16 f32: row (fixed K) striped across 16 lanes (N=lane),
        // same V/half interleave as A.
        v2f b;
        b[0] = W2[(k + 2 * half + 0) * 16 + lr];
        b[1] = W2[(k + 2 * half + 1) * 16 + lr];
        acc = __builtin_amdgcn_wmma_f32_16x16x4_f32(
            false, a, false, b, (short)0, acc, false, false);
    }

    // D 16x16 f32: VGPR v -> M = v + 8*half, N = lr.
    int col = lr;
    if (col < 12) {
        int h = col / 3, t = col - 3 * h;
        float* dst = ws + ((t == 0) ? OFF_ASRC : (t == 1) ? OFF_ADST : OFF_P);
#pragma unroll
        for (int v = 0; v < 8; ++v) {
            int node = tile * 16 + v + 8 * half;
            dst[node * 4 + h] = acc[v];
        }
    }
}

// ---- 3) edge pass 1: segmented max of leaky-relu logits -----------------
__global__ void gat_edge_max(const int* __restrict__ ei, float* __restrict__ ws) {
    int e = blockIdx.x * blockDim.x + threadIdx.x;
    if (e >= N_EDGES) return;
    int s = ei[e];
    int d = ei[N_EDGES + e];
    v4f as = ((const v4f*)(ws + OFF_ASRC))[s];
    v4f ad = ((const v4f*)(ws + OFF_ADST))[d];
    unsigned* mu = (unsigned*)(ws + OFF_MU) + d * 4;
#pragma unroll
    for (int h = 0; h < 4; ++h)
        atomicMax(&mu[h], fp32_orderable(leaky(as[h] + ad[h])));
}

// ---- 4) edge pass 2: exp-sums (denominator + fc-folded numerator) -------
__global__ void gat_edge_sum(const int* __restrict__ ei, float* __restrict__ ws) {
    int e = blockIdx.x * blockDim.x + threadIdx.x;
    if (e >= N_EDGES) return;
    int s = ei[e];
    int d = ei[N_EDGES + e];
    v4f as = ((const v4f*)(ws + OFF_ASRC))[s];
    v4f ad = ((const v4f*)(ws + OFF_ADST))[d];
    v4f ps = ((const v4f*)(ws + OFF_P))[s];
    const unsigned* mu = (const unsigned*)(ws + OFF_MU) + d * 4;
    float* den = ws + OFF_DEN + d * 4;
    float* num = ws + OFF_NUM + d * 4;
#pragma unroll
    for (int h = 0; h < 4; ++h) {
        float m  = fp32_unorderable(mu[h]);
        float ex = __expf(leaky(as[h] + ad[h]) - m);
        atomicAdd(&den[h], ex);
        atomicAdd(&num[h], ex * ps[h]);
    }
}

// ---- 5) finalize: out[n] = sum_h num/(den+eps) + (fc_w.bias + fc_b) -----
__global__ void gat_final(const float* __restrict__ ws, float* __restrict__ out) {
    int n = blockIdx.x * blockDim.x + threadIdx.x;
    if (n >= N_NODES) return;
    v4f den = ((const v4f*)(ws + OFF_DEN))[n];
    v4f num = ((const v4f*)(ws + OFF_NUM))[n];
    float acc = ws[OFF_CC];
#pragma unroll
    for (int h = 0; h < 4; ++h)
        acc += num[h] / (den[h] + 1e-16f);
    out[n] = acc;
}

extern "C" void kernel_launch(void* const* d_in, const int* in_sizes, int n_in,
                              void* d_out, int out_size, void* d_ws, size_t ws_size,
                              hipStream_t stream) {
    const float* x       = (const float*)d_in[0];
    const int*   ei      = (const int*)d_in[1];   // [2, E] int32
    const float* lin_w   = (const float*)d_in[2];
    const float* att_src = (const float*)d_in[3];
    const float* att_dst = (const float*)d_in[4];
    const float* bias    = (const float*)d_in[5];
    const float* fc_w    = (const float*)d_in[6];
    const float* fc_b    = (const float*)d_in[7];
    float* out = (float*)d_out;
    float* ws  = (float*)d_ws;

    (void)in_sizes; (void)n_in; (void)out_size; (void)ws_size;

    gat_fold_w<<<1, 256, 0, stream>>>(lin_w, att_src, att_dst, bias, fc_w, fc_b, ws);
    gat_init<<<(N_NODES * 4 + 255) / 256, 256, 0, stream>>>(ws);

    const int NT = N_NODES / 16;                      // 6250 wave-tiles
    gat_proj_wmma<<<(NT + 3) / 4, 128, 0, stream>>>(x, ws);

    gat_edge_max<<<(N_EDGES + 255) / 256, 256, 0, stream>>>(ei, ws);
    gat_edge_sum<<<(N_EDGES + 255) / 256, 256, 0, stream>>>(ei, ws);
    gat_final<<<(N_NODES + 255) / 256, 256, 0, stream>>>(ws, out);
}